// SimpleGCN_37048387895611
// MI455X (gfx1250) — compile-verified
//
#include <hip/hip_runtime.h>
#include <hip/hip_bf16.h>

#define N_USERS 100000
#define N_ITEMS 50000
#define N_NODES (N_USERS + N_ITEMS)
#define DIM 64
#define N_EDGES 1200000
#define BATCH 4096

typedef float v2f __attribute__((ext_vector_type(2)));
typedef float v8f __attribute__((ext_vector_type(8)));

// ---------------------------------------------------------------------------
// Utility kernels
// ---------------------------------------------------------------------------

__global__ void zero_f32(float* __restrict__ p, size_t n) {
    size_t i = (size_t)blockIdx.x * blockDim.x + threadIdx.x;
    size_t stride = (size_t)gridDim.x * blockDim.x;
    for (; i < n; i += stride) p[i] = 0.0f;
}

__global__ void concat_emb(const float* __restrict__ ue,
                           const float* __restrict__ ie,
                           float* __restrict__ dst) {
    size_t i = (size_t)blockIdx.x * blockDim.x + threadIdx.x;
    const size_t nu = (size_t)N_USERS * DIM;
    const size_t nt = (size_t)N_NODES * DIM;
    if (i < nt) {
        dst[i] = (i < nu) ? ue[i] : ie[i - nu];
    }
}

// deg[row] += 1 for every edge (rows are edge_index[0])
__global__ void deg_count(const int* __restrict__ ei, float* __restrict__ deg) {
    int e = blockIdx.x * blockDim.x + threadIdx.x;
    if (e < N_EDGES) {
        int r = ei[e];
        unsafeAtomicAdd(&deg[r], 1.0f);
    }
}

__global__ void deg_to_dinv(float* __restrict__ deg) {
    int i = blockIdx.x * blockDim.x + threadIdx.x;
    if (i < N_NODES) {
        float d = deg[i];
        deg[i] = (d > 0.0f) ? rsqrtf(d) : 0.0f;
    }
}

// ---------------------------------------------------------------------------
// Propagate: one wave32 per edge, 2 floats per lane (DIM=64).
// dst[col] += dinv[row]*dinv[col] * src[row]
// ---------------------------------------------------------------------------

__global__ void prop_kernel(const int* __restrict__ ei,
                            const float* __restrict__ dinv,
                            const float* __restrict__ src,
                            float* __restrict__ dst) {
    int gtid = blockIdx.x * blockDim.x + threadIdx.x;
    int e    = gtid >> 5;
    int lane = threadIdx.x & 31;
    if (e >= N_EDGES) return;
    int r = ei[e];
    int c = ei[N_EDGES + e];
    float norm = dinv[r] * dinv[c];
    const float2* sp = (const float2*)(src + (size_t)r * DIM);
    float2 v = sp[lane];
    float* dp = dst + (size_t)c * DIM + 2 * lane;
    unsafeAtomicAdd(dp + 0, norm * v.x);
    unsafeAtomicAdd(dp + 1, norm * v.y);
}

// ---------------------------------------------------------------------------
// Dense layer: Y = act(X @ W^T + b), X:[nrows,64], W:[64,64] row-major.
// f32 WMMA 16x16x4, accumulated over K=64 in 16 steps.
// Block = 128 threads = 4 waves; block handles a 16-row tile of X,
// wave w computes output columns [16w, 16w+16).
//
// A 16x4 f32 layout (ISA 7.12.2): lanes 0-15 hold K=kk+0/kk+1 in the two
// VGPRs, lanes 16-31 hold K=kk+2/kk+3. B 4x16 assumed symmetric half-wave
// K split, N striped across lanes. C/D: VGPR v -> M=v (lanes 0-15) or
// M=v+8 (lanes 16-31), N = lane&15.
// ---------------------------------------------------------------------------

template <bool RELU>
__global__ void linear_wmma(const float* __restrict__ X,
                            const float* __restrict__ W,
                            const float* __restrict__ bias,
                            float* __restrict__ Y,
                            int nrows) {
    const int wave = threadIdx.x >> 5;
    const int lane = threadIdx.x & 31;
    const int half = lane >> 4;    // 0 for lanes 0-15, 1 for lanes 16-31
    const int l    = lane & 15;
    const int m0   = blockIdx.x * 16;
    const int n0   = wave * 16;
    if (m0 >= nrows) return;

    v8f c = {};  // f32 accumulator, 8 VGPRs

    const float* xrow = X + (size_t)(m0 + l) * DIM + 2 * half;
    const float* wrow = W + (size_t)(n0 + l) * DIM + 2 * half;

#pragma unroll
    for (int kk = 0; kk < DIM; kk += 4) {
        v2f a, b;
        a.x = xrow[kk + 0];
        a.y = xrow[kk + 1];
        b.x = wrow[kk + 0];
        b.y = wrow[kk + 1];
        // D = A(16x4) * B(4x16) + C ; 8-arg form:
        // (neg_a, A, neg_b, B, c_mod, C, reuse_a, reuse_b)
        c = __builtin_amdgcn_wmma_f32_16x16x4_f32(
                false, a, false, b, (short)0, c, false, false);
    }

    const int ncol = n0 + l;
    const float bv = bias[ncol];
#pragma unroll
    for (int v = 0; v < 8; ++v) {
        int mrow = m0 + v + 8 * half;
        float val = c[v] + bv;
        if (RELU) val = fmaxf(val, 0.0f);
        Y[(size_t)mrow * DIM + ncol] = val;
    }
}

// ---------------------------------------------------------------------------
// Scoring: one wave32 per (user,item) pair, dot over 64 dims.
// ---------------------------------------------------------------------------

__global__ void score_kernel(const int* __restrict__ users,
                             const int* __restrict__ items,
                             const float* __restrict__ emb,
                             float* __restrict__ out) {
    int gtid = blockIdx.x * blockDim.x + threadIdx.x;
    int pair = gtid >> 5;
    int lane = threadIdx.x & 31;
    if (pair >= BATCH) return;
    int u = users[pair];
    int it = items[pair] + N_USERS;
    const float2* up = (const float2*)(emb + (size_t)u * DIM);
    const float2* ip = (const float2*)(emb + (size_t)it * DIM);
    float2 uv = up[lane];
    float2 iv = ip[lane];
    float s = uv.x * iv.x + uv.y * iv.y;
#pragma unroll
    for (int off = 16; off > 0; off >>= 1)
        s += __shfl_xor(s, off, 32);
    if (lane == 0) out[pair] = s;
}

// ---------------------------------------------------------------------------
// Launch
// ---------------------------------------------------------------------------

extern "C" void kernel_launch(void* const* d_in, const int* in_sizes, int n_in,
                              void* d_out, int out_size, void* d_ws, size_t ws_size,
                              hipStream_t stream) {
    const int*   users    = (const int*)d_in[0];
    const int*   items    = (const int*)d_in[1];
    const int*   ei       = (const int*)d_in[2];
    const float* user_emb = (const float*)d_in[3];
    const float* item_emb = (const float*)d_in[4];
    const float* W1       = (const float*)d_in[5];
    const float* b1       = (const float*)d_in[6];
    const float* W2       = (const float*)d_in[7];
    const float* b2       = (const float*)d_in[8];
    float*       out      = (float*)d_out;

    const size_t embN = (size_t)N_NODES * DIM;  // 9.6M floats
    float* buf0 = (float*)d_ws;                 // ping
    float* bufA = buf0 + embN;                  // pong
    float* dinv = bufA + embN;                  // N_NODES floats

    const int T = 256;
    const int embBlocks  = (int)((embN + T - 1) / T);
    const int edgeBlocks = (N_EDGES + T - 1) / T;
    const int nodeBlocks = (N_NODES + T - 1) / T;
    const int waveEdgeBlocks = (int)(((size_t)N_EDGES * 32 + T - 1) / T);
    const int gemmBlocks = N_NODES / 16;        // 150000/16 = 9375 exact
    const int scoreBlocks = (BATCH * 32 + T - 1) / T;

    // buf0 = [user_emb ; item_emb]
    concat_emb<<<embBlocks, T, 0, stream>>>(user_emb, item_emb, buf0);

    // dinv = deg(edge rows)^-0.5
    zero_f32<<<nodeBlocks, T, 0, stream>>>(dinv, (size_t)N_NODES);
    deg_count<<<edgeBlocks, T, 0, stream>>>(ei, dinv);
    deg_to_dinv<<<nodeBlocks, T, 0, stream>>>(dinv);

    // propagate #1 : buf0 -> bufA
    zero_f32<<<embBlocks, T, 0, stream>>>(bufA, embN);
    prop_kernel<<<waveEdgeBlocks, T, 0, stream>>>(ei, dinv, buf0, bufA);

    // linear #1 + relu : bufA -> buf0
    linear_wmma<true><<<gemmBlocks, 128, 0, stream>>>(bufA, W1, b1, buf0, N_NODES);

    // propagate #2 : buf0 -> bufA
    zero_f32<<<embBlocks, T, 0, stream>>>(bufA, embN);
    prop_kernel<<<waveEdgeBlocks, T, 0, stream>>>(ei, dinv, buf0, bufA);

    // linear #2 : bufA -> buf0
    linear_wmma<false><<<gemmBlocks, 128, 0, stream>>>(bufA, W2, b2, buf0, N_NODES);

    // scores
    score_kernel<<<scoreBlocks, T, 0, stream>>>(users, items, buf0, out);
}